// GaussianSmoother_61108794687977
// MI455X (gfx1250) — compile-verified
//
#include <hip/hip_runtime.h>

#define C_ 3
#define D_ 160
#define H_ 192
#define W_ 160
#define S_ (D_ * H_ * W_)   // 4,915,200 voxels per channel

// ---- gfx1250 async global->LDS path (guarded; fallback = plain LDS stores) ----
#if defined(__has_builtin)
# if __has_builtin(__builtin_amdgcn_global_load_async_to_lds_b32) && \
     __has_builtin(__builtin_amdgcn_s_wait_asynccnt)
#  define USE_ASYNC_LDS 1
# endif
#endif
#ifndef USE_ASYNC_LDS
# define USE_ASYNC_LDS 0
#endif

typedef __attribute__((address_space(1))) int gint_t;  // global int
typedef __attribute__((address_space(3))) int lint_t;  // LDS int

// ---- monotonic float <-> ordered-uint mapping (works for negatives) ----
__device__ __forceinline__ unsigned f2ord(float f) {
  unsigned b = __float_as_uint(f);
  return (b & 0x80000000u) ? ~b : (b | 0x80000000u);
}
__device__ __forceinline__ float ord2f(unsigned u) {
  unsigned b = (u & 0x80000000u) ? (u & 0x7fffffffu) : ~u;
  return __uint_as_float(b);
}

// Pass 2: per-voxel std selection + 27-tap separable-Gaussian stencil.
// One block = one 16x16x16 output tile of one channel; LDS holds the 18^3
// padded field tile (23.3 KB), staged via async global->LDS.
// (First in the file so the disasm snippet shows this kernel's body.)
__global__ __launch_bounds__(256) void gs_smooth(
    const float* __restrict__ field, const float* __restrict__ logp,
    float* __restrict__ out, const unsigned* __restrict__ ctrl) {
  __shared__ float tile[18 * 18 * 18];

  const int tx = threadIdx.x, ty = threadIdx.y;
  const int tid = ty * 16 + tx;
  const int tz = blockIdx.z % (D_ / 16);
  const int c  = blockIdx.z / (D_ / 16);
  const int x0 = blockIdx.x * 16, y0 = blockIdx.y * 16, z0 = tz * 16;
  const float* fieldc = field + (size_t)c * S_;

  // Thresholds & the 3 candidate tap weights (uniform across the block).
  const float gmn = ord2f(ctrl[0]), gmx = ord2f(ctrl[1]);
  const float mn = __expf(0.5f * gmn), mx = __expf(0.5f * gmx);
  const float s1 = (2.f * mn + mx) * (1.f / 3.f);
  const float s2 = (mn + 2.f * mx) * (1.f / 3.f);
  const float w1 = __expf(-0.5f / (s1 * s1));
  const float w2 = __expf(-0.5f / (s2 * s2));
  const float w3 = __expf(-0.5f / (mx * mx));
  const float n1 = 1.f + 2.f * w1, n2 = 1.f + 2.f * w2, n3 = 1.f + 2.f * w3;
  const float inv1 = 1.f / (n1 * n1 * n1);
  const float inv2 = 1.f / (n2 * n2 * n2);
  const float inv3 = 1.f / (n3 * n3 * n3);

  // ---- stage 18x18x18 padded tile into LDS ----
  for (int p = tid; p < 18 * 18 * 18; p += 256) {
    const int lx = p % 18;
    const int t  = p / 18;
    const int ly = t % 18;
    const int lz = t / 18;
    const int gx = x0 + lx - 1, gy = y0 + ly - 1, gz = z0 + lz - 1;
    const bool ok = (unsigned)gx < (unsigned)W_ && (unsigned)gy < (unsigned)H_ &&
                    (unsigned)gz < (unsigned)D_;
#if USE_ASYNC_LDS
    if (ok) {
      float* src = const_cast<float*>(fieldc) + (((size_t)gz * H_ + gy) * W_ + gx);
      __builtin_amdgcn_global_load_async_to_lds_b32(
          (gint_t*)src, (lint_t*)&tile[p], 0, 0);
    } else {
      tile[p] = 0.f;  // zero halo; disjoint LDS addresses vs async writes
    }
#else
    tile[p] = ok ? fieldc[((size_t)gz * H_ + gy) * W_ + gx] : 0.f;
#endif
  }
#if USE_ASYNC_LDS
  __builtin_amdgcn_s_wait_asynccnt(0);
#endif
  __syncthreads();

  // ---- compute: each thread owns one (x,y) column, marches 16 z's.
  // Unroll z by 2 so the compiler can CSE the 18 shared LDS operands
  // between consecutive z's (27+27 loads -> ~36).
  const int gx = x0 + tx, gy = y0 + ty;
  const int lx = tx + 1, ly = ty + 1;
#pragma unroll 2
  for (int z = 0; z < 16; ++z) {
    const int gz = z0 + z;
    const int vidx = (gz * H_ + gy) * W_ + gx;
    const float m = fmaxf(fmaxf(logp[vidx], logp[vidx + S_]), logp[vidx + 2 * S_]);
    const float ln = __expf(0.5f * m);
    float w, inv;
    if (ln <= s1)      { w = w1; inv = inv1; }
    else if (ln <= s2) { w = w2; inv = inv2; }
    else               { w = w3; inv = inv3; }
    // tap weight for |dz|+|dy|+|dx| = e is w^e / (1+2w)^3
    float wp[4];
    wp[0] = inv; wp[1] = w * wp[0]; wp[2] = w * wp[1]; wp[3] = w * wp[2];

    const int lz = z + 1;
    float sum = 0.f;
#pragma unroll
    for (int dz = -1; dz <= 1; ++dz)
#pragma unroll
      for (int dy = -1; dy <= 1; ++dy)
#pragma unroll
        for (int dx = -1; dx <= 1; ++dx) {
          const int e = (dz != 0) + (dy != 0) + (dx != 0);
          sum += wp[e] * tile[((lz + dz) * 18 + (ly + dy)) * 18 + (lx + dx)];
        }
    out[(size_t)c * S_ + vidx] = sum;
  }
}

// Pass 1: global min/max of m(v) = max_c log[c][v].  (ln = exp(m/2) later; exp
// is monotonic so reducing m is equivalent and needs one exp at the end.)
__global__ __launch_bounds__(256) void gs_reduce_minmax(
    const float* __restrict__ logp, unsigned* __restrict__ ctrl) {
  __shared__ unsigned smin, smax;
  if (threadIdx.x == 0) { smin = 0xFFFFFFFFu; smax = 0u; }
  __syncthreads();
  unsigned lmin = 0xFFFFFFFFu, lmax = 0u;
  for (int i = blockIdx.x * blockDim.x + threadIdx.x; i < S_;
       i += gridDim.x * blockDim.x) {
    float m = fmaxf(fmaxf(logp[i], logp[i + S_]), logp[i + 2 * S_]);
    unsigned u = f2ord(m);
    lmin = lmin < u ? lmin : u;
    lmax = lmax > u ? lmax : u;
  }
  atomicMin(&smin, lmin);
  atomicMax(&smax, lmax);
  __syncthreads();
  if (threadIdx.x == 0) {
    atomicMin(&ctrl[0], smin);
    atomicMax(&ctrl[1], smax);
  }
}

__global__ void gs_init_ctrl(unsigned* ctrl) {
  ctrl[0] = 0xFFFFFFFFu;  // running min (ordered space)
  ctrl[1] = 0x00000000u;  // running max (ordered space)
}

extern "C" void kernel_launch(void* const* d_in, const int* in_sizes, int n_in,
                              void* d_out, int out_size, void* d_ws, size_t ws_size,
                              hipStream_t stream) {
  const float* field = (const float*)d_in[0];   // [3,160,192,160] f32
  const float* logp  = (const float*)d_in[1];   // [3,160,192,160] f32
  float* out = (float*)d_out;                   // [3,160,192,160] f32
  unsigned* ctrl = (unsigned*)d_ws;             // 2 ordered-uint accumulators

  gs_init_ctrl<<<1, 1, 0, stream>>>(ctrl);
  gs_reduce_minmax<<<1200, 256, 0, stream>>>(logp, ctrl);

  dim3 grid(W_ / 16, H_ / 16, (D_ / 16) * C_);  // 10 x 12 x 30
  dim3 blk(16, 16, 1);                          // 256 threads = 8 waves (wave32)
  gs_smooth<<<grid, blk, 0, stream>>>(field, logp, out, ctrl);
}